// RClassifier0_58978490908736
// MI455X (gfx1250) — compile-verified
//
#include <hip/hip_runtime.h>
#include <hip/hip_bf16.h>

// ---------------------------------------------------------------------------
// Network dims
// ---------------------------------------------------------------------------
#define BATCH 32
#define NCLS  20
// NODE = [211722, 65536, 16384, 4096, 1024, 128]; CH = [1,8,16,32,64,128]; R=8

typedef float v8f  __attribute__((ext_vector_type(8)));
typedef float v2f  __attribute__((ext_vector_type(2)));

// ---------------------------------------------------------------------------
// Fused region-FGL layer: segment_sum over contiguous child ranges + per-node
// (COUT x CIN) region matmul + bias.  Each block handles NPB output nodes.
//   z   : [BATCH][CIN][nin]   (row-major)
//   out : [BATCH][COUT][nout] (row-major)
// parents[i] = i*nout/nin  =>  children(j) = [ceil(j*nin/nout), ceil((j+1)*nin/nout))
// LDS agg laid out [c][n][jl] so phase-1 writes, phase-2 reads and global
// stores are all contiguous in the fastest-varying thread index (jl).
// ---------------------------------------------------------------------------
template <int CIN, int COUT, int NPB>
__global__ void fgl_kernel(const float* __restrict__ z,
                           const float* __restrict__ W,      // [8][COUT][CIN]
                           const float* __restrict__ bias,   // [COUT]
                           const int*   __restrict__ regions,// [nout]
                           float* __restrict__ out,
                           int nin, int nout)
{
    __shared__ float agg[CIN * BATCH * NPB];   // [(c*BATCH + n)*NPB + jl]
    __shared__ int   rsh[NPB];

    const int j0 = blockIdx.x * NPB;

    if (threadIdx.x < NPB)
        rsh[threadIdx.x] = regions[j0 + threadIdx.x];

    // Phase 1: segment-sum children into LDS. Adjacent threads (jl fastest)
    // consume adjacent child ranges of the same input row -> full line use.
    for (int t = threadIdx.x; t < NPB * BATCH * CIN; t += blockDim.x) {
        const int jl = t & (NPB - 1);
        const int c  = (t / NPB) % CIN;
        const int n  = t / (NPB * CIN);
        const long long j = j0 + jl;
        const int s = (int)((j * nin + nout - 1) / nout);
        const int e = (int)(((j + 1) * nin + nout - 1) / nout);   // exclusive
        const float* __restrict__ src = z + ((size_t)n * CIN + c) * (size_t)nin;
        float acc = 0.f;
        for (int i = s; i < e; ++i) acc += src[i];
        agg[(c * BATCH + n) * NPB + jl] = acc;
    }
    __syncthreads();

    // Phase 2: rank-CIN update per (n, d, jl); stores contiguous in j.
    for (int t = threadIdx.x; t < NPB * BATCH * COUT; t += blockDim.x) {
        const int jl = t & (NPB - 1);
        const int d  = (t / NPB) % COUT;
        const int n  = t / (NPB * COUT);
        const float* __restrict__ wr = W + ((size_t)rsh[jl] * COUT + d) * CIN;
        float acc = bias[d];
#pragma unroll
        for (int c = 0; c < CIN; ++c)
            acc += agg[(c * BATCH + n) * NPB + jl] * wr[c];
        out[((size_t)n * COUT + d) * (size_t)nout + j0 + jl] = acc;
    }
}

// ---------------------------------------------------------------------------
// fc1 partial GEMM: hp[y][32][128] = flat[32][Kslice(y)] @ fc1_w[...]^T
// WMMA f32 16x16x4 (exact fp32 semantics, f32 accumulate).
// Grid: (8 N-tiles) x (4 K-split blocks). Block: 256 thr = 8 waves =
// (2 M-halves) x (4 K-chunks of 1024)  => 16-way K parallelism overall.
// K loop: outer 32-wide super-chunk (uniform/scalar prefetch branch) +
// fully unrolled inner 8x WMMA so loads clause and pipeline.
// Bias is added later in fc2 (once), partials reduced there deterministically.
// ---------------------------------------------------------------------------
__global__ void fc1_wmma_kernel(const float* __restrict__ A,    // [32][16384]
                                const float* __restrict__ Wt,   // [128][16384]
                                float* __restrict__ hp)         // [4][32][128]
{
    constexpr int K  = 16384;
    constexpr int KC = 1024;                 // per-wave K chunk
    __shared__ float red[4][2][16][16];      // [kchunk][mhalf][m][n]

    const int lane  = threadIdx.x & 31;
    const int wave  = threadIdx.x >> 5;      // 0..7
    const int mh    = wave & 1;              // M half: rows [mh*16, mh*16+16)
    const int kc    = wave >> 1;             // K chunk within block, 0..3
    const int ntile = blockIdx.x;            // 0..7
    const int kblk  = blockIdx.y;            // 0..3
    const int hi    = lane >> 4;             // lane half selects K pair {0,1}/{2,3}
    const int lm    = lane & 15;

    const int row = mh * 16 + lm;            // A-matrix row (M) held by this lane
    const int col = ntile * 16 + lm;         // B-matrix column (N) held by this lane
    const float* __restrict__ arow = A  + (size_t)row * K;
    const float* __restrict__ bcol = Wt + (size_t)col * K;   // B[k][col] = Wt[col*K + k]
    const float* __restrict__ pa = arow + hi * 2;            // K = ka, ka+1 per lane
    const float* __restrict__ pb = bcol + hi * 2;

    v8f c = {};
    const int kbase = (kblk * 4 + kc) * KC;

    for (int k0 = kbase; k0 < kbase + KC; k0 += 32) {
        if (((k0 & 255) == 0) && (k0 + 1024 + 32 <= K)) { // uniform scalar branch
            __builtin_prefetch(arow + k0 + 1024, 0, 3);   // near-cache prefetch
            __builtin_prefetch(bcol + k0 + 1024, 0, 3);
        }
#pragma unroll
        for (int k = 0; k < 32; k += 4) {
            v2f a;  a.x = pa[k0 + k];  a.y = pa[k0 + k + 1];
            v2f b;  b.x = pb[k0 + k];  b.y = pb[k0 + k + 1];
            c = __builtin_amdgcn_wmma_f32_16x16x4_f32(
                    /*neg_a=*/false, a, /*neg_b=*/false, b,
                    /*c_mod=*/(short)0, c, /*reuse_a=*/false, /*reuse_b=*/false);
        }
    }

    // C/D layout: VGPR v -> M = v + hi*8, N = lm
#pragma unroll
    for (int v = 0; v < 8; ++v)
        red[kc][mh][v + hi * 8][lm] = c[v];
    __syncthreads();

    // Reduce the 4 intra-block K-chunks, store K-split partial
    for (int t = threadIdx.x; t < 2 * 16 * 16; t += blockDim.x) {
        const int m = t >> 4;                // 0..31
        const int n = t & 15;
        const float s = red[0][m >> 4][m & 15][n] + red[1][m >> 4][m & 15][n]
                      + red[2][m >> 4][m & 15][n] + red[3][m >> 4][m & 15][n];
        hp[((size_t)kblk * BATCH + m) * 128 + ntile * 16 + n] = s;
    }
}

// ---------------------------------------------------------------------------
// fc2 (+ deterministic fc1 partial reduce + fc1 bias):
//   h[n][c] = sum_y hp[y][n][c] + fc1_b[c]
//   out[n][o] = sum_c h[n][c] * fc2_w[o][c] + fc2_b[o]
// ---------------------------------------------------------------------------
__global__ void fc2_kernel(const float* __restrict__ hp,    // [4][32][128]
                           const float* __restrict__ fc1_b, // [128]
                           const float* __restrict__ w,     // [20][128]
                           const float* __restrict__ b,     // [20]
                           float* __restrict__ out)         // [32][20]
{
    const int t = blockIdx.x * blockDim.x + threadIdx.x;
    if (t >= BATCH * NCLS) return;
    const int n = t / NCLS, o = t % NCLS;
    float acc = b[o];
#pragma unroll 4
    for (int c = 0; c < 128; ++c) {
        const float hv = hp[(size_t)0 * 4096 + n * 128 + c]
                       + hp[(size_t)1 * 4096 + n * 128 + c]
                       + hp[(size_t)2 * 4096 + n * 128 + c]
                       + hp[(size_t)3 * 4096 + n * 128 + c]
                       + fc1_b[c];
        acc += hv * w[o * 128 + c];
    }
    out[n * NCLS + o] = acc;
}

// ---------------------------------------------------------------------------
// Host launcher
// ---------------------------------------------------------------------------
extern "C" void kernel_launch(void* const* d_in, const int* in_sizes, int n_in,
                              void* d_out, int out_size, void* d_ws, size_t ws_size,
                              hipStream_t stream)
{
    (void)in_sizes; (void)n_in; (void)out_size; (void)ws_size;

    const float* x    = (const float*)d_in[0];
    // per-layer: parents (unused, recomputed arithmetically), regions, W, b
    const int*   reg0 = (const int*)d_in[2];
    const float* W0 = (const float*)d_in[3];  const float* b0 = (const float*)d_in[4];
    const int*   reg1 = (const int*)d_in[6];
    const float* W1 = (const float*)d_in[7];  const float* b1 = (const float*)d_in[8];
    const int*   reg2 = (const int*)d_in[10];
    const float* W2 = (const float*)d_in[11]; const float* b2 = (const float*)d_in[12];
    const int*   reg3 = (const int*)d_in[14];
    const float* W3 = (const float*)d_in[15]; const float* b3 = (const float*)d_in[16];
    const int*   reg4 = (const int*)d_in[18];
    const float* W4 = (const float*)d_in[19]; const float* b4 = (const float*)d_in[20];
    const float* fc1_w = (const float*)d_in[21]; const float* fc1_b = (const float*)d_in[22];
    const float* fc2_w = (const float*)d_in[23]; const float* fc2_b = (const float*)d_in[24];
    float* out = (float*)d_out;

    // Workspace layout (floats)
    float* ws = (float*)d_ws;
    float* z1 = ws;                              // 32*  8*65536 = 16777216
    float* z2 = z1 + (size_t)32 *   8 * 65536;   // 32* 16*16384 =  8388608
    float* z3 = z2 + (size_t)32 *  16 * 16384;   // 32* 32* 4096 =  4194304
    float* z4 = z3 + (size_t)32 *  32 *  4096;   // 32* 64* 1024 =  2097152
    float* z5 = z4 + (size_t)32 *  64 *  1024;   // 32*128*  128 =   524288  (== flat)
    float* hp = z5 + (size_t)32 * 128 *   128;   // 4*32*128 K-split partials

    fgl_kernel<  1,   8, 64><<< 65536/64, 256, 0, stream>>>(x,  W0, b0, reg0, z1, 211722, 65536);
    fgl_kernel<  8,  16, 16><<< 16384/16, 256, 0, stream>>>(z1, W1, b1, reg1, z2,  65536, 16384);
    fgl_kernel< 16,  32,  8><<<  4096/8,  256, 0, stream>>>(z2, W2, b2, reg2, z3,  16384,  4096);
    fgl_kernel< 32,  64,  4><<<  1024/4,  256, 0, stream>>>(z3, W3, b3, reg3, z4,   4096,  1024);
    fgl_kernel< 64, 128,  2><<<   128/2,  256, 0, stream>>>(z4, W4, b4, reg4, z5,   1024,   128);

    fc1_wmma_kernel<<<dim3(8, 4), 256, 0, stream>>>(z5, fc1_w, hp);
    fc2_kernel<<<(BATCH * NCLS + 255) / 256, 256, 0, stream>>>(hp, fc1_b, fc2_w, fc2_b, out);
}